// Fusion_basic_32023276159241
// MI455X (gfx1250) — compile-verified
//
#include <hip/hip_runtime.h>
#include <hip/hip_bf16.h>

#define BATCH 8
#define CHAN  256
#define H     192
#define W     192
#define MH    48
#define MW    48
#define PLANE (H * W)          // 36864 elements, 147456 bytes
#define TBL_STRIDE (6 * H)     // per-batch: y0,y1,x0,x1 (int), wy,wx (float bits)

#if defined(__has_builtin)
#if __has_builtin(__builtin_amdgcn_global_load_async_to_lds_b128)
#define USE_ASYNC_LDS 1
#endif
#endif

typedef int v4i __attribute__((ext_vector_type(4)));
typedef __attribute__((address_space(1))) v4i* gas_v4i_t;
typedef __attribute__((address_space(3))) v4i* lds_v4i_t;

// ---------------------------------------------------------------------------
// Kernel A: bilinear upsample local_map (B,1,48,48) -> lm (B,192,192)
// Exact replica of _axis_static / _resize_bilinear (scale = 48/192 = 0.25).
// ---------------------------------------------------------------------------
__global__ void resize_mask_kernel(const float* __restrict__ local_map,
                                   float* __restrict__ lm) {
    const int b = blockIdx.x;
    const float* m = local_map + b * (MH * MW);
    float* o = lm + b * PLANE;
    for (int e = threadIdx.x; e < PLANE; e += blockDim.x) {
        const int y = e / W;
        const int x = e - y * W;
        float sy = ((float)y + 0.5f) * 0.25f - 0.5f;
        sy = fmaxf(sy, 0.0f);
        int y0 = min((int)floorf(sy), MH - 1);
        int y1 = min(y0 + 1, MH - 1);
        const float wy = sy - (float)y0;
        float sx = ((float)x + 0.5f) * 0.25f - 0.5f;
        sx = fmaxf(sx, 0.0f);
        int x0 = min((int)floorf(sx), MW - 1);
        int x1 = min(x0 + 1, MW - 1);
        const float wx = sx - (float)x0;
        const float tl = m[y0 * MW + x0];
        const float tr = m[y0 * MW + x1];
        const float bl = m[y1 * MW + x0];
        const float br = m[y1 * MW + x1];
        o[e] = tl * (1.0f - wy) * (1.0f - wx) + tr * (1.0f - wy) * wx +
               bl * wy * (1.0f - wx) + br * wy * wx;
    }
}

// ---------------------------------------------------------------------------
// Kernel B: per-batch bbox (exact _bbox semantics, argmax(all-false)==0
// defaults) + precompute _axis_dynamic index/weight tables (192 entries/axis).
// One block per batch, 192 threads.
// ---------------------------------------------------------------------------
__global__ void bbox_axes_kernel(const float* __restrict__ lm,
                                 int* __restrict__ tbl) {
    const int b = blockIdx.x;
    const float* m = lm + b * PLANE;
    __shared__ int row_any[H];
    __shared__ int col_any[W];
    __shared__ int sbox[4];
    const int t = threadIdx.x;

    if (t < H) {
        int ra = 0;
        const float* row = m + t * W;
        #pragma unroll 1
        for (int x = 0; x < W; ++x) ra |= (row[x] != 0.0f) ? 1 : 0;
        row_any[t] = ra;
        int ca = 0;
        #pragma unroll 1
        for (int y = 0; y < H; ++y) ca |= (m[y * W + t] != 0.0f) ? 1 : 0;
        col_any[t] = ca;
    }
    __syncthreads();

    if (t == 0) {
        int min_r = 0;
        #pragma unroll 1
        for (int i = 0; i < H; ++i) { if (row_any[i]) { min_r = i; break; } }
        int max_r = H - 1;
        #pragma unroll 1
        for (int i = H - 1; i >= 0; --i) { if (row_any[i]) { max_r = i; break; } }
        int min_c = 0;
        #pragma unroll 1
        for (int i = 0; i < W; ++i) { if (col_any[i]) { min_c = i; break; } }
        int first_c = 0;
        {
            const float* row = m + min_r * W;
            #pragma unroll 1
            for (int i = 0; i < W; ++i) { if (row[i] != 0.0f) { first_c = i; break; } }
        }
        int last_c = W - 1;
        {
            const float* row = m + max_r * W;
            #pragma unroll 1
            for (int i = W - 1; i >= 0; --i) { if (row[i] != 0.0f) { last_c = i; break; } }
        }
        int nh = max_r - min_r + 1;
        int nw = last_c - first_c + 1;
        nh = min(nh, W - min_r);   // reference clamps with w (==h here)
        nw = min(nw, W - min_c);
        sbox[0] = min_r; sbox[1] = min_c; sbox[2] = nh; sbox[3] = nw;
    }
    __syncthreads();

    if (t < H) {
        const int min_r = sbox[0], min_c = sbox[1], nh = sbox[2], nw = sbox[3];
        int* base = tbl + b * TBL_STRIDE;
        {   // y axis (_axis_dynamic with n=nh, offset=min_r)
            const float scale = (float)nh / (float)H;
            float src = ((float)t + 0.5f) * scale - 0.5f;
            src = fmaxf(src, 0.0f);
            int i0 = min((int)floorf(src), nh - 1);
            int i1 = min(i0 + 1, nh - 1);
            base[0 * H + t] = min_r + i0;
            base[1 * H + t] = min_r + i1;
            base[4 * H + t] = __float_as_int(src - (float)i0);
        }
        {   // x axis (_axis_dynamic with n=nw, offset=min_c)
            const float scale = (float)nw / (float)W;
            float src = ((float)t + 0.5f) * scale - 0.5f;
            src = fmaxf(src, 0.0f);
            int i0 = min((int)floorf(src), nw - 1);
            int i1 = min(i0 + 1, nw - 1);
            base[2 * H + t] = min_c + i0;
            base[3 * H + t] = min_c + i1;
            base[5 * H + t] = __float_as_int(src - (float)i0);
        }
    }
}

// ---------------------------------------------------------------------------
// Kernel C: per-(batch,channel) plane crop-resize. Whole 144KB input plane is
// async-DMA'd into LDS (CDNA5 GLOBAL_LOAD_ASYNC_TO_LDS_B128 + s_wait_asynccnt),
// all bilinear gathers come from LDS, outputs stored as float4 (B128 stores,
// 512B per wave-store). 152KB static LDS -> 2 workgroups per 320KB WGP.
// ---------------------------------------------------------------------------
__global__ __launch_bounds__(256) void crop_resize_kernel(
    const float* __restrict__ f_b, const int* __restrict__ tbl,
    float* __restrict__ out) {
    __shared__ float tile[PLANE];          // 147456 B
    __shared__ int   sy0[H], sy1[H], sx0[W], sx1[W];
    __shared__ float swy[H], swx[W];

    const int bc = blockIdx.x;             // 0 .. BATCH*CHAN-1
    const int b  = bc >> 8;                // CHAN == 256
    const float* plane  = f_b + (size_t)bc * PLANE;
    float*       oplane = out + (size_t)bc * PLANE;
    const int t = threadIdx.x;

    // Kick off the async DMA of the whole plane into LDS first, so it overlaps
    // the table loads below.
#ifdef USE_ASYNC_LDS
    {
        const float4* g = (const float4*)plane;
        float4*       l = (float4*)tile;
        for (int i = t; i < PLANE / 4; i += 256) {
            __builtin_amdgcn_global_load_async_to_lds_b128(
                (gas_v4i_t)(g + i), (lds_v4i_t)(l + i), 0, 0);
        }
    }
#else
    {
        const float4* g = (const float4*)plane;
        float4*       l = (float4*)tile;
        for (int i = t; i < PLANE / 4; i += 256) l[i] = g[i];
    }
#endif

    // Load per-batch interpolation tables into LDS (regular loads, overlap DMA).
    const int* base = tbl + b * TBL_STRIDE;
    if (t < H) {
        sy0[t] = base[0 * H + t];
        sy1[t] = base[1 * H + t];
        sx0[t] = base[2 * H + t];
        sx1[t] = base[3 * H + t];
        swy[t] = __int_as_float(base[4 * H + t]);
        swx[t] = __int_as_float(base[5 * H + t]);
    }

#ifdef USE_ASYNC_LDS
#if defined(__has_builtin) && __has_builtin(__builtin_amdgcn_s_wait_asynccnt)
    __builtin_amdgcn_s_wait_asynccnt(0);
#else
    asm volatile("s_wait_asynccnt 0" ::: "memory");
#endif
#endif
    __syncthreads();

    // 9216 float4 output groups / 256 threads = 36 per thread. Each row of 192
    // is exactly 48 groups, so a group never straddles a row: y-table lookup is
    // hoisted, stores are B128 and fully coalesced.
    for (int g = t; g < PLANE / 4; g += 256) {
        const int y  = g / (W / 4);
        const int xg = g - y * (W / 4);
        const int x  = xg * 4;
        const int y0 = sy0[y], y1 = sy1[y];
        const float wy = swy[y];
        const float omwy = 1.0f - wy;
        const float* row0 = &tile[y0 * W];
        const float* row1 = &tile[y1 * W];
        float4 r;
        float* rp = (float*)&r;
        #pragma unroll
        for (int k = 0; k < 4; ++k) {
            const int x0 = sx0[x + k], x1 = sx1[x + k];
            const float wx = swx[x + k];
            const float omwx = 1.0f - wx;
            const float tl = row0[x0];
            const float tr = row0[x1];
            const float bl = row1[x0];
            const float br = row1[x1];
            rp[k] = tl * omwy * omwx + tr * omwy * wx +
                    bl * wy * omwx + br * wy * wx;
        }
        ((float4*)oplane)[g] = r;
    }
}

// ---------------------------------------------------------------------------
extern "C" void kernel_launch(void* const* d_in, const int* in_sizes, int n_in,
                              void* d_out, int out_size, void* d_ws, size_t ws_size,
                              hipStream_t stream) {
    const float* f_b       = (const float*)d_in[0];   // (8,256,192,192) fp32
    const float* local_map = (const float*)d_in[1];   // (8,1,48,48) fp32
    float* out = (float*)d_out;                       // (8,256,192,192) fp32

    // Workspace layout: [ lm : B*PLANE floats ][ tables : B*6*H ints/floats ]
    float* lm  = (float*)d_ws;
    int*   tbl = (int*)d_ws + BATCH * PLANE;

    resize_mask_kernel<<<BATCH, 256, 0, stream>>>(local_map, lm);
    bbox_axes_kernel<<<BATCH, 192, 0, stream>>>(lm, tbl);
    crop_resize_kernel<<<BATCH * CHAN, 256, 0, stream>>>(f_b, tbl, out);
}